// MambaClassifier_58729382805602
// MI455X (gfx1250) — compile-verified
//
#include <hip/hip_runtime.h>
#include <hip/hip_bf16.h>

typedef __attribute__((ext_vector_type(2))) float v2f;
typedef __attribute__((ext_vector_type(8))) float v8f;

// ---- problem constants (match reference) ----
static constexpr int B_   = 16;
static constexpr int L_   = 1024;
static constexpr int DM   = 256;   // d_model
static constexpr int DI   = 512;   // d_inner
static constexpr int DS   = 16;    // d_state
static constexpr int DR   = 16;    // dt_rank
static constexpr int IN_F = 64;
static constexpr int OUT_D= 128;
static constexpr int MID_ = 50;
static constexpr int NCLS = 3;
static constexpr int M_   = B_ * L_;           // 16384 rows
static constexpr int DBCW = DR + 2 * DS;       // 48

// ---- workspace layout (floats) ----
static constexpr size_t OFF_H     = 0;                      // [M, 256]
static constexpr size_t OFF_HN    = OFF_H  + (size_t)M_*DM; // [M, 256]
static constexpr size_t OFF_XR    = OFF_HN + (size_t)M_*DM; // [M, 1024] (also head out)
static constexpr size_t OFF_XS    = OFF_XR + (size_t)M_*2*DI;// [M, 512]
static constexpr size_t OFF_DBC   = OFF_XS + (size_t)M_*DI; // [M, 48]
static constexpr size_t OFF_DELTA = OFF_DBC+ (size_t)M_*DBCW;// [M, 512] (also y)
static constexpr size_t OFF_FEAT  = OFF_DELTA + (size_t)M_*DI; // [16,128]

__device__ __forceinline__ float siluf(float x) { return x / (1.f + __builtin_expf(-x)); }
__device__ __forceinline__ float softplusf(float x) {
  return (x > 20.f) ? x : log1pf(__builtin_expf(x));
}
__device__ __forceinline__ v2f ld2(const float* __restrict__ p) {
  return *(const v2f*)p;   // 8B-aligned by construction (even strides/offsets)
}

// =====================================================================
// Generic fp32 WMMA GEMM:  C[M, ldc] (cols n0..) = A[M, lda] * W[N, K]^T
// One wave computes a 16 x (16*NT) tile with V_WMMA_F32_16X16X4_F32.
// Fragment layouts per CDNA5 ISA 7.12.2 (f32 16x16x4):
//   A: lane&15 = m, lane>>4 selects K-pair {2h, 2h+1} -> contiguous float2
//   B: lane&15 = n, lane>>4 selects K-pair {2h, 2h+1} -> contiguous float2 of W row
//   C/D: col = lane&15, row = vgpr_j + 8*(lane>>4)
// K-loop is register double-buffered: fragment loads for step k+4 are issued
// as one clause before the WMMAs of step k, so s_wait_loadcnt overlaps with
// matrix math instead of draining to 0 before every WMMA.
// =====================================================================
template<int NT, bool BIAS, int ACT, bool ACCUM>
__global__ __launch_bounds__(32)
void wmma_gemm_kernel(const float* __restrict__ A, int lda,
                      const float* __restrict__ W,
                      const float* __restrict__ bias,
                      float* __restrict__ C, int ldc, int K) {
  const int lane = threadIdx.x;
  const int half = lane >> 4;
  const int r    = lane & 15;
  const int m0 = blockIdx.x * 16;
  const int n0 = blockIdx.y * (16 * NT);

  v8f acc[NT];
#pragma unroll
  for (int t = 0; t < NT; ++t)
#pragma unroll
    for (int j = 0; j < 8; ++j) acc[t][j] = 0.f;

  const float* ap = A + (size_t)(m0 + r) * lda + 2 * half;
  const float* wp = W + (size_t)(n0 + r) * K + 2 * half;

  // prologue: fragments for k = 0
  v2f a_cur = ld2(ap);
  v2f b_cur[NT];
#pragma unroll
  for (int t = 0; t < NT; ++t) b_cur[t] = ld2(wp + (size_t)(t * 16) * K);

#pragma unroll 2
  for (int k0 = 4; k0 < K; k0 += 4) {
    // issue next-step fragment loads first (one clause, stays outstanding)
    v2f a_nxt = ld2(ap + k0);
    v2f b_nxt[NT];
#pragma unroll
    for (int t = 0; t < NT; ++t) b_nxt[t] = ld2(wp + (size_t)(t * 16) * K + k0);
    // matrix math on current fragments overlaps the loads above
#pragma unroll
    for (int t = 0; t < NT; ++t)
      acc[t] = __builtin_amdgcn_wmma_f32_16x16x4_f32(false, a_cur, false, b_cur[t],
                                                     (short)0, acc[t], false, false);
    a_cur = a_nxt;
#pragma unroll
    for (int t = 0; t < NT; ++t) b_cur[t] = b_nxt[t];
  }
  // epilogue step
#pragma unroll
  for (int t = 0; t < NT; ++t)
    acc[t] = __builtin_amdgcn_wmma_f32_16x16x4_f32(false, a_cur, false, b_cur[t],
                                                   (short)0, acc[t], false, false);

#pragma unroll
  for (int t = 0; t < NT; ++t) {
    const int n = n0 + t * 16 + r;
    const float bv = BIAS ? bias[n] : 0.f;
#pragma unroll
    for (int j = 0; j < 8; ++j) {
      const int m = m0 + j + 8 * half;
      float v = acc[t][j] + bv;
      if (ACT == 1) v = softplusf(v);
      float* cp = C + (size_t)m * ldc + n;
      if (ACCUM) *cp += v; else *cp = v;
    }
  }
}

// ---- RMSNorm over rows of width 256: one wave per row ----
__global__ __launch_bounds__(256)
void rmsnorm_kernel(const float* __restrict__ x, const float* __restrict__ w,
                    float* __restrict__ o, int rows) {
  const int wid  = (int)((blockIdx.x * blockDim.x + threadIdx.x) >> 5);
  const int lane = threadIdx.x & 31;
  if (wid >= rows) return;
  const float* xr = x + (size_t)wid * DM;
  float v[8];
  float s = 0.f;
#pragma unroll
  for (int j = 0; j < 8; ++j) { v[j] = xr[lane + 32 * j]; s += v[j] * v[j]; }
#pragma unroll
  for (int off = 16; off; off >>= 1) s += __shfl_xor(s, off, 32);
  const float scale = rsqrtf(s * (1.0f / (float)DM) + 1e-5f);
  float* orow = o + (size_t)wid * DM;
#pragma unroll
  for (int j = 0; j < 8; ++j) {
    const int c = lane + 32 * j;
    orow[c] = v[j] * scale * w[c];
  }
}

// ---- causal depthwise conv (K=4) over xm = xr[:, 0:512], then SiLU ----
__global__ __launch_bounds__(256)
void conv_silu_kernel(const float* __restrict__ xr, const float* __restrict__ wc,
                      const float* __restrict__ cb, float* __restrict__ xs) {
  const size_t idx = (size_t)blockIdx.x * blockDim.x + threadIdx.x; // (b*L+l)*DI + d
  if (idx >= (size_t)M_ * DI) return;
  const int d = (int)(idx & (DI - 1));
  const size_t bl = idx >> 9;          // b*L + l
  const int l = (int)(bl & (L_ - 1));
  const float* w4 = wc + d * 4;
  float acc = cb[d];
#pragma unroll
  for (int k = 0; k < 4; ++k) {
    const int ls = l - 3 + k;
    if (ls >= 0) acc += xr[(bl - 3 + k) * (2 * DI) + d] * w4[k];
  }
  xs[idx] = siluf(acc);
}

// ---- selective scan: thread per (b,d); state h[16] + A row in registers ----
__global__ __launch_bounds__(128)
void scan_kernel(const float* __restrict__ delta, const float* __restrict__ xs,
                 const float* __restrict__ dbc, const float* __restrict__ xr,
                 const float* __restrict__ A_log, const float* __restrict__ Dp,
                 float* __restrict__ yout) {
  const int b = blockIdx.x >> 2;                    // DI/128 = 4 chunks
  const int d = ((blockIdx.x & 3) * 128) + threadIdx.x;
  float Aexp[DS], hst[DS];
#pragma unroll
  for (int n = 0; n < DS; ++n) {
    Aexp[n] = -__builtin_expf(A_log[d * DS + n]);
    hst[n] = 0.f;
  }
  const float Dpd = Dp[d];
  const size_t base = (size_t)b * L_;
  for (int l = 0; l < L_; ++l) {
    const size_t bl = base + l;
    const float dl = delta[bl * DI + d];
    const float xv = xs[bl * DI + d];
    const float* row = dbc + bl * DBCW;             // [dlt16 | B16 | C16]
    float y = 0.f;
#pragma unroll
    for (int n = 0; n < DS; ++n) {
      const float dA = __builtin_expf(dl * Aexp[n]);
      hst[n] = dA * hst[n] + (dl * row[DR + n]) * xv;
      y += hst[n] * row[DR + DS + n];
    }
    const float res = xr[bl * (2 * DI) + DI + d];
    yout[bl * DI + d] = (y + xv * Dpd) * siluf(res); // yout aliases delta: safe (read first)
  }
}

// ---- mean over L: one wave per (b, o) ----
__global__ __launch_bounds__(256)
void mean_kernel(const float* __restrict__ ho, float* __restrict__ feat) {
  const int gid = blockIdx.x * blockDim.x + threadIdx.x;
  const int wid = gid >> 5, lane = gid & 31;
  if (wid >= B_ * OUT_D) return;
  const int b = wid >> 7, o = wid & (OUT_D - 1);
  const float* p = ho + (size_t)b * L_ * OUT_D + o;
  float s = 0.f;
  for (int l = lane; l < L_; l += 32) s += p[(size_t)l * OUT_D];
#pragma unroll
  for (int off = 16; off; off >>= 1) s += __shfl_xor(s, off, 32);
  if (lane == 0) feat[wid] = s * (1.f / (float)L_);
}

// ---- tiny classifier head MLP: [16,128] -> relu[16,50] -> [16,3] ----
__global__ __launch_bounds__(256)
void mlp_kernel(const float* __restrict__ feat,
                const float* __restrict__ W1, const float* __restrict__ b1,
                const float* __restrict__ W2, const float* __restrict__ b2,
                float* __restrict__ out) {
  __shared__ float hid[B_ * MID_];
  const int tid = threadIdx.x;
  for (int idx = tid; idx < B_ * MID_; idx += blockDim.x) {
    const int b = idx / MID_, m = idx % MID_;
    float s = b1[m];
    const float* f = feat + b * OUT_D;
    const float* w = W1 + m * OUT_D;
    for (int k = 0; k < OUT_D; ++k) s += f[k] * w[k];
    hid[idx] = fmaxf(s, 0.f);
  }
  __syncthreads();
  for (int idx = tid; idx < B_ * NCLS; idx += blockDim.x) {
    const int b = idx / NCLS, c = idx % NCLS;
    float s = b2[c];
    for (int k = 0; k < MID_; ++k) s += hid[b * MID_ + k] * W2[c * MID_ + k];
    out[idx] = s;
  }
}

extern "C" void kernel_launch(void* const* d_in, const int* in_sizes, int n_in,
                              void* d_out, int out_size, void* d_ws, size_t ws_size,
                              hipStream_t stream) {
  const float* x       = (const float*)d_in[0];
  const float* emb_W   = (const float*)d_in[1];
  const float* emb_b   = (const float*)d_in[2];
  const float* in_W    = (const float*)d_in[3];
  const float* conv_W  = (const float*)d_in[4];
  const float* conv_b  = (const float*)d_in[5];
  const float* xp_W    = (const float*)d_in[6];
  const float* dt_W    = (const float*)d_in[7];
  const float* dt_b    = (const float*)d_in[8];
  const float* A_log   = (const float*)d_in[9];
  const float* Dp      = (const float*)d_in[10];
  const float* out_W   = (const float*)d_in[11];
  const float* norm_W  = (const float*)d_in[12];
  const float* normf_W = (const float*)d_in[13];
  const float* head_W  = (const float*)d_in[14];
  const float* head_b  = (const float*)d_in[15];
  const float* W1      = (const float*)d_in[16];
  const float* b1      = (const float*)d_in[17];
  const float* W2      = (const float*)d_in[18];
  const float* b2      = (const float*)d_in[19];
  float* out = (float*)d_out;
  float* ws  = (float*)d_ws;

  float* h     = ws + OFF_H;
  float* hn    = ws + OFF_HN;
  float* xr    = ws + OFF_XR;
  float* xs    = ws + OFF_XS;
  float* dbc   = ws + OFF_DBC;
  float* delta = ws + OFF_DELTA;   // reused as y after the scan
  float* feat  = ws + OFF_FEAT;
  float* ho    = xr;               // head output reuses xr space (free by then)

  // h = x @ emb_W^T + emb_b                        [16384,64] -> [16384,256]
  wmma_gemm_kernel<4, true, 0, false><<<dim3(M_ / 16, DM / 64), 32, 0, stream>>>(
      x, IN_F, emb_W, emb_b, h, DM, IN_F);

  for (int i = 0; i < 4; ++i) {
    // hn = rmsnorm(h, norm_W[i])
    rmsnorm_kernel<<<M_ / 8, 256, 0, stream>>>(h, norm_W + (size_t)i * DM, hn, M_);
    // xr = hn @ in_W[i]^T                          [16384,256] -> [16384,1024]
    wmma_gemm_kernel<4, false, 0, false><<<dim3(M_ / 16, (2 * DI) / 64), 32, 0, stream>>>(
        hn, DM, in_W + (size_t)i * 2 * DI * DM, nullptr, xr, 2 * DI, DM);
    // xs = silu(causal_dwconv(xm) + conv_b)
    conv_silu_kernel<<<(M_ * DI) / 256, 256, 0, stream>>>(
        xr, conv_W + (size_t)i * DI * 4, conv_b + (size_t)i * DI, xs);
    // dbc = xs @ xp_W[i]^T                         [16384,512] -> [16384,48]
    wmma_gemm_kernel<3, false, 0, false><<<dim3(M_ / 16, 1), 32, 0, stream>>>(
        xs, DI, xp_W + (size_t)i * DBCW * DI, nullptr, dbc, DBCW, DI);
    // delta = softplus(dlt @ dt_W[i]^T + dt_b[i])  [16384,16] -> [16384,512]
    wmma_gemm_kernel<4, true, 1, false><<<dim3(M_ / 16, DI / 64), 32, 0, stream>>>(
        dbc, DBCW, dt_W + (size_t)i * DI * DR, dt_b + (size_t)i * DI, delta, DI, DR);
    // y = (scan(...) + xs*Dp) * silu(res)   (y written in-place over delta)
    scan_kernel<<<B_ * (DI / 128), 128, 0, stream>>>(
        delta, xs, dbc, xr, A_log + (size_t)i * DI * DS, Dp + (size_t)i * DI, delta);
    // h += y @ out_W[i]^T                          [16384,512] -> [16384,256]
    wmma_gemm_kernel<4, false, 0, true><<<dim3(M_ / 16, DM / 64), 32, 0, stream>>>(
        delta, DI, out_W + (size_t)i * DM * DI, nullptr, h, DM, DI);
  }

  // final norm + head + mean + MLP
  rmsnorm_kernel<<<M_ / 8, 256, 0, stream>>>(h, normf_W, hn, M_);
  wmma_gemm_kernel<4, true, 0, false><<<dim3(M_ / 16, OUT_D / 64), 32, 0, stream>>>(
      hn, DM, head_W, head_b, ho, OUT_D, DM);
  mean_kernel<<<(B_ * OUT_D * 32) / 256, 256, 0, stream>>>(ho, feat);
  mlp_kernel<<<1, 256, 0, stream>>>(feat, W1, b1, W2, b2, out);
}